// VDP_Conv2D_26499948216839
// MI455X (gfx1250) — compile-verified
//
#include <hip/hip_runtime.h>

typedef float v2f __attribute__((ext_vector_type(2)));
typedef float v4f __attribute__((ext_vector_type(4)));
typedef float v8f __attribute__((ext_vector_type(8)));

#define IC_   3
#define H_    32
#define W_    32
#define OH_   28
#define OW_   28
#define L_    784      // 28*28
#define F_    75       // 3*5*5
#define OC_   32
#define B_    4
#define NT_   49       // 784 / 16 tiles
#define KSTEPS 19      // ceil(75/4)

// X[f, l] = mu[b, ic, oh+kh, ow+kw] with f = ic*25 + kh*5 + kw, l = oh*28 + ow.
__device__ __forceinline__ float patchX(const float* __restrict__ muB, int f, int l) {
    int ic = f / 25;
    int r  = f - ic * 25;
    int kh = r / 5;
    int kw = r - kh * 5;
    int oh = l / OW_;
    int ow = l - oh * OW_;
    return muB[ic * (H_ * W_) + (oh + kh) * W_ + (ow + kw)];
}

// f32 16x16x4 fragment (A 16x4 and B 4x16 share this layout):
//   lane = idx + 16*(k>=2), vgpr = k&1   ->  element (idx, k)
__device__ __forceinline__ v2f load_fragX(const float* __restrict__ muB, int kstep, int lbase) {
    int lane  = threadIdx.x & 31;
    int idx   = lane & 15;
    int khalf = lane >> 4;
    int k0    = kstep * 4 + khalf * 2;
    int l     = lbase + idx;
    v2f r;
    r.x = (k0     < F_) ? patchX(muB, k0,     l) : 0.0f;
    r.y = (k0 + 1 < F_) ? patchX(muB, k0 + 1, l) : 0.0f;
    return r;
}

__device__ __forceinline__ v2f load_fragW(const float* __restrict__ Wt, int kstep, int ocbase) {
    int lane  = threadIdx.x & 31;
    int idx   = lane & 15;
    int khalf = lane >> 4;
    int k0    = kstep * 4 + khalf * 2;
    const float* wrow = Wt + (size_t)(ocbase + idx) * F_;
    v2f r;
    r.x = (k0     < F_) ? wrow[k0]     : 0.0f;
    r.y = (k0 + 1 < F_) ? wrow[k0 + 1] : 0.0f;
    return r;
}

// ---------------- mu_z = W @ X + b via WMMA ----------------
// One wave per 16x16 output tile: 4 batches * 2 oc-tiles * 49 l-tiles = 392 waves.
__global__ __launch_bounds__(256)
void conv_wmma_kernel(const float* __restrict__ mu, const float* __restrict__ Wt,
                      const float* __restrict__ bias, float* __restrict__ mu_z) {
    int wave = (blockIdx.x * blockDim.x + threadIdx.x) >> 5;
    if (wave >= B_ * 2 * NT_) return;
    int tl  = wave % NT_;
    int t2  = wave / NT_;
    int toc = t2 % 2;
    int b   = t2 / 2;
    const float* muB = mu + (size_t)b * IC_ * H_ * W_;
    int oc0 = toc * 16;
    int l1  = tl * 16;

    v8f c = {};
#pragma unroll
    for (int ks = 0; ks < KSTEPS; ++ks) {
        v2f a  = load_fragW(Wt,  ks, oc0);
        v2f bf = load_fragX(muB, ks, l1);
        c = __builtin_amdgcn_wmma_f32_16x16x4_f32(false, a, false, bf, (short)0, c,
                                                  false, false);
    }
    int lane = threadIdx.x & 31;
    int n    = lane & 15;
    int hi   = lane >> 4;
    float* outB = mu_z + (size_t)b * OC_ * L_;
#pragma unroll
    for (int p = 0; p < 8; ++p) {
        int oc = oc0 + p + hi * 8;
        outB[(size_t)oc * L_ + l1 + n] = c[p] + bias[oc];
    }
}

// ---------------- Gram tiles via WMMA ----------------
// One wave per 16x16 tile of x_matrix: 4 * 49 * 49 = 9604 waves.
// FUSED=false: write x_matrix[b,l,m] to ws.  FUSED=true: write sigma_z directly.
template <bool FUSED>
__global__ __launch_bounds__(256)
void gram_wmma_kernel(const float* __restrict__ mu, const float* __restrict__ sigma_w,
                      const float* __restrict__ sigma_b, float* __restrict__ out) {
    int wave = (blockIdx.x * blockDim.x + threadIdx.x) >> 5;
    if (wave >= B_ * NT_ * NT_) return;
    int tm = wave % NT_;
    int t2 = wave / NT_;
    int tl = t2 % NT_;
    int b  = t2 / NT_;
    const float* muB = mu + (size_t)b * IC_ * H_ * W_;
    int l0 = tl * 16;   // row block
    int l1 = tm * 16;   // col block

    v8f c = {};
#pragma unroll
    for (int ks = 0; ks < KSTEPS; ++ks) {
        v2f a  = load_fragX(muB, ks, l0);   // A[m,k] = X[k, l0+m]
        v2f bf = load_fragX(muB, ks, l1);   // B[k,n] = X[k, l1+n]
        c = __builtin_amdgcn_wmma_f32_16x16x4_f32(false, a, false, bf, (short)0, c,
                                                  false, false);
    }
    int lane = threadIdx.x & 31;
    int n    = lane & 15;
    int hi   = lane >> 4;
    if (FUSED) {
        float sb = sigma_b[0];
#pragma unroll 1
        for (int oc = 0; oc < OC_; ++oc) {
            float sp = log1pf(expf(sigma_w[oc]));
            float* dst = out + (size_t)(b * OC_ + oc) * L_ * L_;
#pragma unroll
            for (int p = 0; p < 8; ++p) {
                int row = l0 + p + hi * 8;
                __builtin_nontemporal_store(sp * c[p] + sb,
                                            &dst[(size_t)row * L_ + l1 + n]);
            }
        }
    } else {
        float* xm = out + (size_t)b * L_ * L_;
#pragma unroll
        for (int p = 0; p < 8; ++p) {
            int row = l0 + p + hi * 8;
            xm[(size_t)row * L_ + l1 + n] = c[p];
        }
    }
}

// ---------------- sigma_z broadcast: pure store bandwidth ----------------
// One float4 along m per thread; NT stores keep x_matrix resident in L2.
__global__ __launch_bounds__(256)
void sigma_bcast_kernel(const float* __restrict__ xm, const float* __restrict__ sigma_w,
                        const float* __restrict__ sigma_b, float* __restrict__ sigma_z) {
    int j = blockIdx.x * blockDim.x + threadIdx.x;  // [0, B*OC*L*196)
    const int N4 = B_ * OC_ * L_ * (L_ / 4);
    if (j >= N4) return;
    int m0  = (j % (L_ / 4)) * 4;
    int row = j / (L_ / 4);         // row = (b*OC + oc)*L + l
    int l   = row % L_;
    int boc = row / L_;
    int oc  = boc & (OC_ - 1);
    int b   = boc >> 5;

    const v4f xv = *(const v4f*)(xm + ((size_t)b * L_ + l) * L_ + m0);
    float sp = log1pf(expf(sigma_w[oc]));
    float sb = sigma_b[0];
    v4f o;
    o.x = sp * xv.x + sb;
    o.y = sp * xv.y + sb;
    o.z = sp * xv.z + sb;
    o.w = sp * xv.w + sb;
    __builtin_nontemporal_store(o, (v4f*)(sigma_z + (size_t)row * L_ + m0));
}

extern "C" void kernel_launch(void* const* d_in, const int* in_sizes, int n_in,
                              void* d_out, int out_size, void* d_ws, size_t ws_size,
                              hipStream_t stream) {
    const float* mu      = (const float*)d_in[0];
    const float* Wt      = (const float*)d_in[1];
    const float* bias    = (const float*)d_in[2];
    const float* sigma_w = (const float*)d_in[3];
    const float* sigma_b = (const float*)d_in[4];

    float* out     = (float*)d_out;
    float* mu_z    = out;                           // [4,32,28,28] = 100352
    float* sigma_z = out + (size_t)B_ * OC_ * L_;   // [4,32,784,784]

    // mu_z conv (392 waves)
    {
        int waves  = B_ * 2 * NT_;
        int blocks = (waves * 32 + 255) / 256;
        conv_wmma_kernel<<<blocks, 256, 0, stream>>>(mu, Wt, bias, mu_z);
    }

    int gwaves  = B_ * NT_ * NT_;                   // 9604
    int gblocks = (gwaves * 32 + 255) / 256;        // 1201

    size_t xm_bytes = (size_t)B_ * L_ * L_ * sizeof(float);  // 9.8 MB
    if (ws_size >= xm_bytes) {
        float* xm = (float*)d_ws;
        gram_wmma_kernel<false><<<gblocks, 256, 0, stream>>>(mu, sigma_w, sigma_b, xm);
        int n4     = B_ * OC_ * L_ * (L_ / 4);      // 19,668,992
        int bblocks = (n4 + 255) / 256;
        sigma_bcast_kernel<<<bblocks, 256, 0, stream>>>(xm, sigma_w, sigma_b, sigma_z);
    } else {
        gram_wmma_kernel<true><<<gblocks, 256, 0, stream>>>(mu, sigma_w, sigma_b, sigma_z);
    }
}